// CompressedSensing_83141976916725
// MI455X (gfx1250) — compile-verified
//
#include <hip/hip_runtime.h>
#include <math.h>

// ---------------------------------------------------------------------------
// FISTA compressed sensing on MI455X (gfx1250), wave32.
//  - V_WMMA_F32_16X16X4_F32 for both GEMMs (f32 precision; ~86 GFLOP total,
//    L2/latency bound -> fp16 throughput buys nothing, precision matters).
//  - 32 persistent blocks, each owning 16 batch rows for all 100 iterations
//    (iteration decouples over batch -> no grid-wide sync ever).
//  - Row-pair-interleaved matp/matT in d_ws: every WMMA B operand is one
//    global_load_b64 (two 128B segments).
//  - y chunks staged transposed into LDS via CDNA5 async global->LDS scatter
//    (global_load_async_to_lds_b32 + s_wait_asynccnt), ALSO pair-interleaved
//    so every WMMA A operand is one naturally-aligned ds_load_b64 (no movs).
// ---------------------------------------------------------------------------

typedef float v2f __attribute__((ext_vector_type(2)));
typedef float v8f __attribute__((ext_vector_type(8)));

#define B_TOT    512
#define N_DIM    5184
#define M_DIM    81
#define M_PAD    96           // 6 WMMA tiles of 16
#define N_ITERS  100
#define BT       16           // batch rows per block (WMMA M)
#define NWAVES   16
#define NTHREADS (NWAVES * 32)
#define KCHUNK   32           // k-values staged per LDS chunk
#define NCHUNKS  (N_DIM / KCHUNK)   // 162
#define NTILES   (N_DIM / 16)       // 324
#define ZSTRIDE  98           // even (b64-aligned), 98%64=34 -> conflict-free
#define YTPSTRIDE 34          // dwords per k-pair block: 16 rows * 2 + 2 pad

// matpP: [2592][96][2]  pair p = matp rows (2p, 2p+1), padded m
// matTP: [48][5184][2]  pair q = matT rows (2q, 2q+1)
__global__ void fista_setup(const float* __restrict__ mat,
                            float* __restrict__ matpP,
                            float* __restrict__ matTP) {
  int idx = blockIdx.x * blockDim.x + threadIdx.x;
  if (idx >= N_DIM * M_PAD) return;
  int n = idx / M_PAD;
  int m = idx - n * M_PAD;
  float v = (m < M_DIM) ? mat[n * M_DIM + m] : 0.0f;
  matpP[(((size_t)(n >> 1)) * M_PAD + m) * 2 + (n & 1)] = v;
  matTP[(((size_t)(m >> 1)) * N_DIM + n) * 2 + (m & 1)] = v;
}

__global__ __launch_bounds__(NTHREADS) void fista_main(
    const float* __restrict__ xf,     // [512][81]
    const float* __restrict__ lam,    // [1]
    const v2f* __restrict__ matpP,    // [2592][96] of float2
    const v2f* __restrict__ matTP,    // [48][5184] of float2
    float* __restrict__ y) {          // [512][5184] = output
  __shared__ __align__(16) float zbuf[BT * ZSTRIDE];                      // z then r
  __shared__ __align__(16) float ytile[NWAVES * (KCHUNK / 2) * YTPSTRIDE]; // per-wave yT

  const int tid  = threadIdx.x;
  const int wave = tid >> 5;
  const int lane = tid & 31;
  const bool low = (lane < 16);
  const int col  = low ? lane : (lane - 16);   // tile column / batch row
  const int rofs = low ? 0 : 2;                // K-row offset for upper lanes
  const int bbase = blockIdx.x * BT;

  float* yblk = y + (size_t)bbase * N_DIM;
  const float thr = lam[0];                    // lam / MU, MU == 1
  float* myYT = &ytile[wave * ((KCHUNK / 2) * YTPSTRIDE)];
  // yT layout: element (k, r) at dword (k>>1)*34 + 2r + (k&1).
  // Per-lane async-scatter base (lane l writes column k = k0 + l):
  // low 32 bits of a flat shared pointer == LDS byte offset (shared aperture)
  const unsigned myYTbase =
      (unsigned)(uintptr_t)(myYT + (lane >> 1) * YTPSTRIDE + (lane & 1));

  const v8f zero8 = {0.f, 0.f, 0.f, 0.f, 0.f, 0.f, 0.f, 0.f};

  // y0 = 0 for this block's tile
  for (int i = tid; i < BT * N_DIM; i += NTHREADS) yblk[i] = 0.0f;
  __threadfence();
  __syncthreads();

  float t = 1.0f;
  for (int it = 0; it < N_ITERS; ++it) {
    // ---- zero z accumulator ----
    for (int i = tid; i < BT * ZSTRIDE; i += NTHREADS) zbuf[i] = 0.0f;
    __syncthreads();

    // ---- phase 1: z[16][96] = y_tile @ matp, K striped over waves ----
    v8f acc1[6];
#pragma unroll
    for (int mt = 0; mt < 6; ++mt) acc1[mt] = zero8;

    for (int ch = wave; ch < NCHUNKS; ch += NWAVES) {
      const int k0 = ch * KCHUNK;
      // async transpose-scatter: lane l brings y[r][k0+l] into yT(k=l, r)
      {
        const float* gb = yblk + k0 + lane;
#pragma unroll
        for (int r = 0; r < BT; ++r) {
          const unsigned lo = myYTbase + 8u * r;
          const unsigned long long ga =
              (unsigned long long)(uintptr_t)(gb + (size_t)r * N_DIM);
          asm volatile("global_load_async_to_lds_b32 %0, %1, off"
                       :: "v"(lo), "v"(ga) : "memory");
        }
        asm volatile("s_wait_asynccnt 0x0" ::: "memory");
      }
#pragma unroll
      for (int kk = 0; kk < KCHUNK / 4; ++kk) {
        // A 16x4: K rows (kb, kb+1) for this lane half; single ds_load_b64
        const v2f a = *(const v2f*)&myYT[(2 * kk + (rofs >> 1)) * YTPSTRIDE +
                                         2 * col];
        const v2f* bp =
            matpP + (size_t)((k0 >> 1) + 2 * kk + (rofs >> 1)) * M_PAD + col;
#pragma unroll
        for (int mt = 0; mt < 6; ++mt) {
          v2f b = bp[mt * 16];       // one b64: rows (k,k+1) interleaved
          acc1[mt] = __builtin_amdgcn_wmma_f32_16x16x4_f32(
              false, a, false, b, (short)0, acc1[mt], false, false);
        }
      }
    }
    // cross-wave reduce via LDS float atomics
#pragma unroll
    for (int mt = 0; mt < 6; ++mt) {
#pragma unroll
      for (int g = 0; g < 8; ++g) {
        const int row = g + (low ? 0 : 8);   // D: VGPR g -> M=g / g+8
        atomicAdd(&zbuf[row * ZSTRIDE + mt * 16 + col], acc1[mt][g]);
      }
    }
    __syncthreads();

    // ---- r = xf - z (in place; padded cols -> 0) ----
    for (int e = tid; e < BT * M_PAD; e += NTHREADS) {
      const int row = e / M_PAD;
      const int c   = e - row * M_PAD;
      float v = 0.0f;
      if (c < M_DIM)
        v = xf[(size_t)(bbase + row) * M_DIM + c] - zbuf[row * ZSTRIDE + c];
      zbuf[row * ZSTRIDE + c] = v;
    }
    __syncthreads();

    // ---- phase 2: re = r @ matT per n-tile + fused FISTA update ----
    const float tn   = 0.5f * (1.0f + sqrtf(1.0f + t * t));
    const float cmom = (t - 1.0f) / tn;

    v2f av[24];                      // A (r): K=96 -> 24 k-steps, b64 loads
#pragma unroll
    for (int kk = 0; kk < 24; ++kk)
      av[kk] = *(const v2f*)&zbuf[col * ZSTRIDE + 4 * kk + rofs];

    for (int nt = wave; nt < NTILES; nt += NWAVES) {
      const int nb = nt * 16;
      v8f acc = zero8;
      const v2f* bp = matTP + (size_t)(rofs >> 1) * N_DIM + nb + col;
#pragma unroll
      for (int kk = 0; kk < 24; ++kk) {
        v2f b = bp[(size_t)(2 * kk) * N_DIM];   // pair q = 2kk + rofs/2
        acc = __builtin_amdgcn_wmma_f32_16x16x4_f32(
            false, av[kk], false, b, (short)0, acc, false, false);
      }
#pragma unroll
      for (int g = 0; g < 8; ++g) {
        const int row = g + (low ? 0 : 8);
        float* yp = yblk + (size_t)row * N_DIM + nb + col;
        const float yo = *yp;
        const float w  = yo + acc[g];                        // MU == 1
        const float yn = copysignf(fmaxf(fabsf(w) - thr, 0.0f), w);
        *yp = yn + cmom * (yn - yo);
      }
    }

    t = tn;
    __threadfence();   // y writes visible to next iteration's reads
    __syncthreads();
  }
}

extern "C" void kernel_launch(void* const* d_in, const int* in_sizes, int n_in,
                              void* d_out, int out_size, void* d_ws, size_t ws_size,
                              hipStream_t stream) {
  const float* x   = (const float*)d_in[0];   // [512][81]
  const float* mat = (const float*)d_in[1];   // [5184][81]
  const float* lam = (const float*)d_in[2];   // [1]
  float* y = (float*)d_out;                   // [512][5184]

  float* matpP = (float*)d_ws;                             // 2592*96*2 f32 ~2MB
  float* matTP = matpP + (size_t)(N_DIM / 2) * M_PAD * 2;  // 48*5184*2 f32 ~2MB

  fista_setup<<<(N_DIM * M_PAD + 255) / 256, 256, 0, stream>>>(mat, matpP, matTP);
  fista_main<<<B_TOT / BT, NTHREADS, 0, stream>>>(
      x, lam, (const v2f*)matpP, (const v2f*)matTP, y);
}